// GATNet_66666482369141
// MI455X (gfx1250) — compile-verified
//
#include <hip/hip_runtime.h>
#include <hip/hip_bf16.h>

// ---------------------------------------------------------------------------
// GATNet (2x GATConv + dot decode) for gfx1250.
// GEMMs via V_WMMA_F32_16X16X4_F32, 5 M-tiles register-blocked per wave32
// (one B fragment feeds 5 WMMAs per K-step).
// Segment softmax via ordered-uint atomicMax + f32 atomicAdd (L2-resident).
// ---------------------------------------------------------------------------

#define NN   30000
#define EE   480000
#define ETOT (EE + NN)
#define INC  128
#define H1   5
#define C1   64
#define F1   (H1 * C1)   // 320
#define OUTC 64
#define PEC  100000
#define NEG_SLOPE 0.2f
#define MB   5           // M-tiles per wave; NN = 375 * (16*MB)

typedef __attribute__((ext_vector_type(2))) float v2f;
typedef __attribute__((ext_vector_type(8))) float v8f;

// ---- float <-> monotone-ordered uint key (for atomicMax over floats) ------
__device__ __forceinline__ unsigned fkey(float f) {
    unsigned u = __float_as_uint(f);
    return (u & 0x80000000u) ? ~u : (u | 0x80000000u);
}
__device__ __forceinline__ float fkey_inv(unsigned k) {
    return (k & 0x80000000u) ? __uint_as_float(k & 0x7FFFFFFFu)
                             : __uint_as_float(~k);
}

__device__ __forceinline__ void get_edge(const int* __restrict__ ei, int e,
                                         int& s, int& d) {
    if (e < EE) { s = ei[e]; d = ei[EE + e]; }
    else        { s = e - EE; d = s; }        // implicit self loops
}

// ---------------------------------------------------------------------------
// fills
// ---------------------------------------------------------------------------
__global__ void fill_f32(float* __restrict__ p, float v, int n) {
    int i = blockIdx.x * blockDim.x + threadIdx.x;
    if (i < n) p[i] = v;
}
__global__ void fill_u32(unsigned* __restrict__ p, unsigned v, int n) {
    int i = blockIdx.x * blockDim.x + threadIdx.x;
    if (i < n) p[i] = v;
}

// ---------------------------------------------------------------------------
// C[M,N] = A[M,K] @ B[K,N], fp32 WMMA. Wave32 computes an 80x16 output strip
// (MB=5 stacked 16x16 tiles) so each B fragment is loaded once per K-step and
// reused by 5 consecutive v_wmma_f32_16x16x4_f32 issues.
// A-frag: lanes 0-15 -> M=lane, v0/v1 = K+0/K+1; lanes 16-31 -> K+2/K+3.
// C: VGPR r -> row r (lanes 0-15) / row r+8 (lanes 16-31), col = lane&15.
// Requires M % (16*MB) == 0, N % 16 == 0, K % 4 == 0 (true for all calls).
// ---------------------------------------------------------------------------
__global__ void gemm_wmma_f32(const float* __restrict__ A,
                              const float* __restrict__ B,
                              float* __restrict__ C,
                              int M, int N, int K) {
    const int lane   = threadIdx.x & 31;
    const int waveId = (blockIdx.x * blockDim.x + threadIdx.x) >> 5;
    const int tilesN = N >> 4;
    const int mblks  = M / (16 * MB);
    if (waveId >= mblks * tilesN) return;  // wave-uniform: EXEC stays all-1s

    const int tm0  = (waveId / tilesN) * MB;
    const int tn   = waveId % tilesN;
    const int half = lane >> 4;            // 0: lanes 0-15, 1: lanes 16-31
    const int l15  = lane & 15;
    const int col  = tn * 16 + l15;

    const float* Arow[MB];
    #pragma unroll
    for (int r = 0; r < MB; ++r)
        Arow[r] = A + (size_t)((tm0 + r) * 16 + l15) * K;

    v8f acc[MB];
    #pragma unroll
    for (int r = 0; r < MB; ++r) acc[r] = (v8f){};

    for (int k = 0; k < K; k += 4) {
        const int k0 = k + (half << 1);
        v2f b;
        b.x = B[(size_t)k0 * N + col];
        b.y = B[(size_t)(k0 + 1) * N + col];
        #pragma unroll
        for (int r = 0; r < MB; ++r) {
            v2f a;
            a.x = Arow[r][k0];
            a.y = Arow[r][k0 + 1];
            acc[r] = __builtin_amdgcn_wmma_f32_16x16x4_f32(
                /*neg_a=*/false, a, /*neg_b=*/false, b,
                /*c_mod=*/(short)0, acc[r],
                /*reuse_a=*/false, /*reuse_b=*/false);
        }
    }

    #pragma unroll
    for (int r = 0; r < MB; ++r) {
        const int rowbase = (tm0 + r) * 16 + (half << 3);
        #pragma unroll
        for (int q = 0; q < 8; ++q)
            C[(size_t)(rowbase + q) * N + col] = acc[r][q];
    }
}

// ---------------------------------------------------------------------------
// a_src[n,h] = sum_c h[n,h*C+c]*att_src[h,c]; a_dst likewise. Wave per (n,h).
// ---------------------------------------------------------------------------
__global__ void att_scores(const float* __restrict__ h,
                           const float* __restrict__ att_src,
                           const float* __restrict__ att_dst,
                           float* __restrict__ a_src,
                           float* __restrict__ a_dst,
                           int n_nodes, int heads, int ch) {
    const int lane = threadIdx.x & 31;
    const int w    = (blockIdx.x * blockDim.x + threadIdx.x) >> 5;
    if (w >= n_nodes * heads) return;
    const int n  = w / heads;
    const int hd = w % heads;
    const float* hp = h + (size_t)n * heads * ch + (size_t)hd * ch;
    const float* as = att_src + (size_t)hd * ch;
    const float* ad = att_dst + (size_t)hd * ch;
    float ss = 0.f, sd = 0.f;
    for (int c = lane; c < ch; c += 32) {
        float v = hp[c];
        ss += v * as[c];
        sd += v * ad[c];
    }
    #pragma unroll
    for (int m = 16; m >= 1; m >>= 1) {
        ss += __shfl_xor(ss, m, 32);
        sd += __shfl_xor(sd, m, 32);
    }
    if (lane == 0) {
        a_src[w] = ss;
        a_dst[w] = sd;
    }
}

// ---------------------------------------------------------------------------
// Pass A: e = leaky_relu(a_src[src]+a_dst[dst]); store e; atomicMax key at dst.
// Thread per (edge, head).
// ---------------------------------------------------------------------------
__global__ void edge_max(const float* __restrict__ a_src,
                         const float* __restrict__ a_dst,
                         const int* __restrict__ ei,
                         unsigned* __restrict__ mkey,
                         float* __restrict__ eval,
                         int heads) {
    const int t = blockIdx.x * blockDim.x + threadIdx.x;
    if (t >= ETOT * heads) return;
    const int e  = t / heads;
    const int hd = t % heads;
    int s, d;
    get_edge(ei, e, s, d);
    float v = a_src[s * heads + hd] + a_dst[d * heads + hd];
    v = (v > 0.f) ? v : NEG_SLOPE * v;
    eval[t] = v;
    atomicMax(&mkey[d * heads + hd], fkey(v));
}

// ---------------------------------------------------------------------------
// Pass B: ex = exp(e - m[dst]); overwrite eval with ex; atomicAdd sum at dst.
// ---------------------------------------------------------------------------
__global__ void edge_expsum(const unsigned* __restrict__ mkey,
                            const int* __restrict__ ei,
                            float* __restrict__ eval,
                            float* __restrict__ ssum,
                            int heads) {
    const int t = blockIdx.x * blockDim.x + threadIdx.x;
    if (t >= ETOT * heads) return;
    const int e  = t / heads;
    const int hd = t % heads;
    int s, d;
    get_edge(ei, e, s, d);
    const float m  = fkey_inv(mkey[d * heads + hd]);
    const float ex = __expf(eval[t] - m);
    eval[t] = ex;
    atomicAdd(&ssum[d * heads + hd], ex);
}

// ---------------------------------------------------------------------------
// Pass C: out[dst, :] += (ex/s[dst]) * h[src, :]. Wave per edge, all H*C ch.
// ---------------------------------------------------------------------------
__global__ void edge_scatter(const float* __restrict__ hfeat,
                             const float* __restrict__ eval,
                             const float* __restrict__ ssum,
                             const int* __restrict__ ei,
                             float* __restrict__ outf,
                             int heads, int ch) {
    const int lane = threadIdx.x & 31;
    const int e    = (blockIdx.x * blockDim.x + threadIdx.x) >> 5;
    if (e >= ETOT) return;
    int s, d;
    get_edge(ei, e, s, d);
    const int F = heads * ch;
    const float* hs = hfeat + (size_t)s * F;
    float*       od = outf  + (size_t)d * F;
    for (int c = lane; c < F; c += 32) {
        const int hd = c / ch;
        const float alpha =
            eval[e * heads + hd] / (ssum[d * heads + hd] + 1e-16f);
        atomicAdd(&od[c], alpha * hs[c]);
    }
}

// ---------------------------------------------------------------------------
// In-place bias (+ optional ReLU) on an [n_rows, ld] accumulator.
// ---------------------------------------------------------------------------
__global__ void bias_act(float* __restrict__ z, const float* __restrict__ b,
                         int n_elems, int ld, int do_relu) {
    const int i = blockIdx.x * blockDim.x + threadIdx.x;
    if (i >= n_elems) return;
    float v = z[i] + b[i % ld];
    if (do_relu) v = v > 0.f ? v : 0.f;
    z[i] = v;
}

// ---------------------------------------------------------------------------
// Decode: out[i] = dot(z[a_i], z[b_i]) over 64 ch. Wave per pair.
// i < PE -> pos edges, else neg edges.
// ---------------------------------------------------------------------------
__global__ void decode_dot(const float* __restrict__ z,
                           const int* __restrict__ pos,
                           const int* __restrict__ neg,
                           float* __restrict__ out) {
    const int lane = threadIdx.x & 31;
    const int i    = (blockIdx.x * blockDim.x + threadIdx.x) >> 5;
    if (i >= 2 * PEC) return;
    int a, b;
    if (i < PEC) { a = pos[i];       b = pos[PEC + i]; }
    else         { a = neg[i - PEC]; b = neg[PEC + (i - PEC)]; }
    const float* za = z + (size_t)a * OUTC;
    const float* zb = z + (size_t)b * OUTC;
    float s = za[lane] * zb[lane] + za[lane + 32] * zb[lane + 32];
    #pragma unroll
    for (int m = 16; m >= 1; m >>= 1) s += __shfl_xor(s, m, 32);
    if (lane == 0) out[i] = s;
}

// ---------------------------------------------------------------------------
// Host launcher
// ---------------------------------------------------------------------------
static inline int ceil_div(long long a, long long b) { return (int)((a + b - 1) / b); }

extern "C" void kernel_launch(void* const* d_in, const int* in_sizes, int n_in,
                              void* d_out, int out_size, void* d_ws, size_t ws_size,
                              hipStream_t stream) {
    const float* x        = (const float*)d_in[0];
    const int*   ei       = (const int*)  d_in[1];
    const int*   pos      = (const int*)  d_in[2];
    const int*   neg      = (const int*)  d_in[3];
    const float* W1       = (const float*)d_in[4];
    const float* att_s1   = (const float*)d_in[5];
    const float* att_d1   = (const float*)d_in[6];
    const float* b1       = (const float*)d_in[7];
    const float* W2       = (const float*)d_in[8];
    const float* att_s2   = (const float*)d_in[9];
    const float* att_d2   = (const float*)d_in[10];
    const float* b2       = (const float*)d_in[11];
    float*       out      = (float*)d_out;

    // workspace carve-out (256B aligned)
    char*  ws  = (char*)d_ws;
    size_t off = 0;
    auto carve = [&](size_t bytes) -> void* {
        void* p = ws + off;
        off = (off + bytes + 255) & ~(size_t)255;
        return p;
    };
    float*    h1    = (float*)   carve((size_t)NN * F1 * 4);
    float*    out1  = (float*)   carve((size_t)NN * F1 * 4);   // becomes z1
    float*    as1   = (float*)   carve((size_t)NN * H1 * 4);
    float*    ad1   = (float*)   carve((size_t)NN * H1 * 4);
    unsigned* mkey1 = (unsigned*)carve((size_t)NN * H1 * 4);
    float*    s1    = (float*)   carve((size_t)NN * H1 * 4);
    float*    ev1   = (float*)   carve((size_t)ETOT * H1 * 4);
    float*    h2    = (float*)   carve((size_t)NN * OUTC * 4);
    float*    out2  = (float*)   carve((size_t)NN * OUTC * 4); // becomes z2
    float*    as2   = (float*)   carve((size_t)NN * 4);
    float*    ad2   = (float*)   carve((size_t)NN * 4);
    unsigned* mkey2 = (unsigned*)carve((size_t)NN * 4);
    float*    s2    = (float*)   carve((size_t)NN * 4);
    float*    ev2   = (float*)   carve((size_t)ETOT * 4);

    const int TB = 256;                 // 8 waves / block
    const int WPB = TB / 32;

    // ---------------- layer 1 ----------------
    fill_f32<<<ceil_div((size_t)NN * F1, TB), TB, 0, stream>>>(out1, 0.f, NN * F1);
    fill_f32<<<ceil_div(NN * H1, TB), TB, 0, stream>>>(s1, 0.f, NN * H1);
    fill_u32<<<ceil_div(NN * H1, TB), TB, 0, stream>>>(mkey1, 0u, NN * H1);

    {   // h1 = x @ W1   (30000x128 * 128x320)
        int waves = (NN / (16 * MB)) * (F1 / 16);
        gemm_wmma_f32<<<ceil_div(waves, WPB), TB, 0, stream>>>(x, W1, h1, NN, F1, INC);
    }
    att_scores<<<ceil_div(NN * H1, WPB), TB, 0, stream>>>(h1, att_s1, att_d1,
                                                          as1, ad1, NN, H1, C1);
    edge_max   <<<ceil_div((size_t)ETOT * H1, TB), TB, 0, stream>>>(as1, ad1, ei, mkey1, ev1, H1);
    edge_expsum<<<ceil_div((size_t)ETOT * H1, TB), TB, 0, stream>>>(mkey1, ei, ev1, s1, H1);
    edge_scatter<<<ceil_div(ETOT, WPB), TB, 0, stream>>>(h1, ev1, s1, ei, out1, H1, C1);
    bias_act<<<ceil_div((size_t)NN * F1, TB), TB, 0, stream>>>(out1, b1, NN * F1, F1, 1);

    // ---------------- layer 2 ----------------
    fill_f32<<<ceil_div((size_t)NN * OUTC, TB), TB, 0, stream>>>(out2, 0.f, NN * OUTC);
    fill_f32<<<ceil_div(NN, TB), TB, 0, stream>>>(s2, 0.f, NN);
    fill_u32<<<ceil_div(NN, TB), TB, 0, stream>>>(mkey2, 0u, NN);

    {   // h2 = z1 @ W2  (30000x320 * 320x64)
        int waves = (NN / (16 * MB)) * (OUTC / 16);
        gemm_wmma_f32<<<ceil_div(waves, WPB), TB, 0, stream>>>(out1, W2, h2, NN, OUTC, F1);
    }
    att_scores<<<ceil_div(NN, WPB), TB, 0, stream>>>(h2, att_s2, att_d2,
                                                     as2, ad2, NN, 1, OUTC);
    edge_max   <<<ceil_div(ETOT, TB), TB, 0, stream>>>(as2, ad2, ei, mkey2, ev2, 1);
    edge_expsum<<<ceil_div(ETOT, TB), TB, 0, stream>>>(mkey2, ei, ev2, s2, 1);
    edge_scatter<<<ceil_div(ETOT, WPB), TB, 0, stream>>>(h2, ev2, s2, ei, out2, 1, OUTC);
    bias_act<<<ceil_div((size_t)NN * OUTC, TB), TB, 0, stream>>>(out2, b2, NN * OUTC, OUTC, 0);

    // ---------------- decode ----------------
    decode_dot<<<ceil_div(2 * PEC, WPB), TB, 0, stream>>>(out2, pos, neg, out);
}